// RiemannianMeanPool_89876485636504
// MI455X (gfx1250) — compile-verified
//
#include <hip/hip_runtime.h>
#include <hip/hip_bf16.h>
#include <math.h>

#define DIM 32
#define DD  1024            // 32*32
#define NMAT 512
#define BATCH 64
#define JITTER 1e-5f
#define TOL 1e-4f

typedef __attribute__((ext_vector_type(2))) float v2f;
typedef __attribute__((ext_vector_type(8))) float v8f;

// ---------------------------------------------------------------------------
// One wave (32 lanes) owns one 32x32 f32 matrix resident in LDS.
// 32x32 matmul = 2x2 tiles of V_WMMA_F32_16X16X4_F32, K=32 -> 8 steps.
// f32 WMMA layouts (ISA 7.12.2):
//   A (16x4): lanes 0-15 rows 0..15 {K=k,k+1}; lanes 16-31 rows 0..15 {K=k+2,k+3}
//   B (4x16): VGPR0: lanes0-15 K=k, lanes16-31 K=k+2; VGPR1: K=k+1 / K=k+3
//   C (16x16): VGPR v: lanes0-15 row v, lanes16-31 row v+8
// ---------------------------------------------------------------------------
__device__ __forceinline__ void mm32(const float* __restrict__ A,
                                     const float* __restrict__ B,
                                     float* __restrict__ C,
                                     int lane, float scale) {
  __syncthreads();
  const int half = lane >> 4;
  const int l16  = lane & 15;
  v8f acc00 = {}, acc01 = {}, acc10 = {}, acc11 = {};
#pragma unroll
  for (int kk = 0; kk < 8; ++kk) {
    const int k0 = kk * 4 + half * 2;
    v2f a0 = *(const v2f*)&A[(l16      ) * DIM + k0];
    v2f a1 = *(const v2f*)&A[(l16 + 16 ) * DIM + k0];
    v2f b0, b1;
    b0.x = B[(k0    ) * DIM + l16];
    b0.y = B[(k0 + 1) * DIM + l16];
    b1.x = B[(k0    ) * DIM + 16 + l16];
    b1.y = B[(k0 + 1) * DIM + 16 + l16];
    acc00 = __builtin_amdgcn_wmma_f32_16x16x4_f32(false, a0, false, b0, (short)0, acc00, false, false);
    acc01 = __builtin_amdgcn_wmma_f32_16x16x4_f32(false, a0, false, b1, (short)0, acc01, false, false);
    acc10 = __builtin_amdgcn_wmma_f32_16x16x4_f32(false, a1, false, b0, (short)0, acc10, false, false);
    acc11 = __builtin_amdgcn_wmma_f32_16x16x4_f32(false, a1, false, b1, (short)0, acc11, false, false);
  }
  __syncthreads();
#pragma unroll
  for (int v = 0; v < 8; ++v) {
    const int r = v + 8 * half;
    C[(r     ) * DIM + l16     ] = acc00[v] * scale;
    C[(r     ) * DIM + 16 + l16] = acc01[v] * scale;
    C[(r + 16) * DIM + l16     ] = acc10[v] * scale;
    C[(r + 16) * DIM + 16 + l16] = acc11[v] * scale;
  }
  __syncthreads();
}

// ---------------- elementwise helpers on LDS-resident matrices --------------
__device__ __forceinline__ void copy_m(float* dst, const float* src, int lane) {
  for (int t = lane; t < DD; t += 32) dst[t] = src[t];
}
__device__ __forceinline__ void set_identity(float* M, int lane) {
  for (int t = lane; t < DD; t += 32) M[t] = ((t >> 5) == (t & 31)) ? 1.0f : 0.0f;
  __syncthreads();
}
__device__ __forceinline__ void three_i_minus(float* T, int lane) {
  for (int t = lane; t < DD; t += 32)
    T[t] = (((t >> 5) == (t & 31)) ? 3.0f : 0.0f) - T[t];
  __syncthreads();
}
__device__ __forceinline__ void scale_m(float* M, float s, int lane) {
  for (int t = lane; t < DD; t += 32) M[t] *= s;
  __syncthreads();
}
__device__ __forceinline__ void add_scaled(float* acc, const float* P, float c, int lane) {
  for (int t = lane; t < DD; t += 32) acc[t] += c * P[t];
  __syncthreads();
}
__device__ __forceinline__ void gload(float* dst, const float* __restrict__ src, int lane) {
#pragma unroll
  for (int k = 0; k < 8; ++k) {
    const int t = lane * 4 + k * 128;
    *(float4*)&dst[t] = *(const float4*)&src[t];
  }
  __syncthreads();
}
__device__ __forceinline__ void gstore_scaled(float* __restrict__ dst, const float* src,
                                              float s, int lane) {
  __syncthreads();
  for (int t = lane; t < DD; t += 32) dst[t] = src[t] * s;
}
__device__ __forceinline__ float trace_mean(const float* A, int lane) {
  __syncthreads();
  float tr = A[lane * 33];            // diag element, one per lane
#pragma unroll
  for (int off = 16; off > 0; off >>= 1) tr += __shfl_xor(tr, off, 32);
  return tr * (1.0f / 32.0f);
}

// Coupled Newton-Schulz: on entry Y holds S with spectrum near 1.
// On exit Y ~= sqrt(S), Z ~= S^{-1/2}. T,U,V scratch.
__device__ __forceinline__ void ns_coupled(float* Y, float* Z, float* T, float* U, float* V,
                                           int lane, int iters) {
  set_identity(Z, lane);
#pragma unroll 1
  for (int it = 0; it < iters; ++it) {
    mm32(Z, Y, T, lane, 1.0f);        // T = Z*Y
    three_i_minus(T, lane);           // T = 3I - Z*Y
    mm32(Y, T, U, lane, 0.5f);        // U = 0.5*Y*T
    mm32(T, Z, V, lane, 0.5f);        // V = 0.5*T*Z
    copy_m(Y, U, lane);
    copy_m(Z, V, lane);
    __syncthreads();
  }
}

// ------------------------------- kernels -----------------------------------
__global__ void init_kernel(float* nrm, int* done) {
  *nrm = 0.0f; *done = 0;
}

__global__ void mean_kernel(const float* __restrict__ X, float* __restrict__ G) {
  const int b = blockIdx.x;
  for (int e = threadIdx.x; e < DD; e += 256) {
    float s = 0.0f;
    const float* xb = X + (size_t)b * NMAT * DD + e;
    for (int n = 0; n < NMAT; ++n) s += xb[(size_t)n * DD];
    G[(size_t)b * DD + e] = s * (1.0f / (float)NMAT);
  }
}

__global__ void zero_kernel(float* __restrict__ Delta, float* nrm) {
  const int t = blockIdx.x * blockDim.x + threadIdx.x;
  if (t < BATCH * DD) Delta[t] = 0.0f;
  if (t == 0) *nrm = 0.0f;
}

// G^{1/2}, G^{-1/2} per batch via trace-normalized coupled Newton-Schulz.
__global__ void __launch_bounds__(32)
sqrt_invsqrt_kernel(const float* __restrict__ G, float* __restrict__ Gs,
                    float* __restrict__ Gis) {
  __shared__ float Y[DD], Z[DD], T[DD], U[DD], V[DD];
  const int b = blockIdx.x, lane = threadIdx.x;
  gload(Y, G + (size_t)b * DD, lane);
  float c = trace_mean(Y, lane);
  c = fmaxf(c, 1e-6f);
  scale_m(Y, 1.0f / c, lane);
  ns_coupled(Y, Z, T, U, V, lane, 12);
  const float sc = sqrtf(c);
  gstore_scaled(Gs  + (size_t)b * DD, Y, sc, lane);
  gstore_scaled(Gis + (size_t)b * DD, Z, 1.0f / sc, lane);
}

// A = Gis*X*Gis ; logA via inverse scaling-squaring (3 NS roots + series);
// atomically accumulate Delta[b] += logA / N.
__global__ void __launch_bounds__(32)
log_accum_kernel(const float* __restrict__ X, const float* __restrict__ Gis,
                 float* __restrict__ Delta) {
  __shared__ float B0[DD], B1[DD], B2[DD], B3[DD], B4[DD], B5[DD], B6[DD];
  const int id = blockIdx.x;            // b*N + n
  const int batch = id >> 9;            // /512
  const int lane = threadIdx.x;
  gload(B0, Gis + (size_t)batch * DD, lane);
  gload(B1, X + (size_t)id * DD, lane);
  mm32(B0, B1, B2, lane, 1.0f);         // Gis*X
  mm32(B2, B0, B1, lane, 1.0f);         // A = Gis*X*Gis
  float logshift = 0.0f;
#pragma unroll 1
  for (int s = 0; s < 3; ++s) {
    float c = trace_mean(B1, lane);
    c = fmaxf(c, 1e-6f);
    scale_m(B1, 1.0f / c, lane);
    logshift += logf(c) * (float)(1 << s);
    copy_m(B2, B1, lane); __syncthreads();          // Y = A
    ns_coupled(B2, B3, B4, B5, B6, lane, 8);        // Y -> sqrt(A)
    copy_m(B1, B2, lane); __syncthreads();          // A = sqrt(A)
  }
  // E = A - I ; log(I+E) ~= E - E^2/2 + ... +- E^6/6
  for (int t = lane; t < DD; t += 32)
    B2[t] = B1[t] - (((t >> 5) == (t & 31)) ? 1.0f : 0.0f);
  __syncthreads();
  copy_m(B3, B2, lane);                 // Epow = E
  copy_m(B4, B2, lane);                 // L    = E
  __syncthreads();
#pragma unroll 1
  for (int m = 2; m <= 6; ++m) {
    mm32(B3, B2, B5, lane, 1.0f);       // Epow *= E
    copy_m(B3, B5, lane); __syncthreads();
    const float coef = ((m & 1) ? 1.0f : -1.0f) / (float)m;
    add_scaled(B4, B3, coef, lane);
  }
  const float outer = 8.0f;             // 2^3 roots
  const float invN  = 1.0f / (float)NMAT;
  for (int t = lane; t < DD; t += 32) {
    const float v = outer * B4[t] + (((t >> 5) == (t & 31)) ? logshift : 0.0f);
    atomicAdd(&Delta[(size_t)batch * DD + t], v * invN);
  }
}

__global__ void __launch_bounds__(32)
norm_kernel(const float* __restrict__ Delta, float* nrm) {
  const int b = blockIdx.x, lane = threadIdx.x;
  float ss = 0.0f;
  for (int t = lane; t < DD; t += 32) {
    const float v = Delta[(size_t)b * DD + t];
    ss += v * v;
  }
#pragma unroll
  for (int off = 16; off > 0; off >>= 1) ss += __shfl_xor(ss, off, 32);
  if (lane == 0) atomicAdd(nrm, sqrtf(ss) * (1.0f / (float)BATCH));
}

// G_new = sym(Gs * expm_taylor(Delta) * Gs) + jitter*I, frozen if done.
__global__ void __launch_bounds__(32)
update_kernel(float* __restrict__ G, const float* __restrict__ Gs,
              const float* __restrict__ Delta, const int* done) {
  if (*done) return;                    // wave-uniform
  __shared__ float XD[DD], S[DD], OUT[DD], P[DD], T[DD];
  const int b = blockIdx.x, lane = threadIdx.x;
  gload(XD, Delta + (size_t)b * DD, lane);
  gload(S,  Gs    + (size_t)b * DD, lane);
  for (int t = lane; t < DD; t += 32) {
    const float x = XD[t];
    P[t] = x;
    OUT[t] = x + (((t >> 5) == (t & 31)) ? 1.0f : 0.0f);   // I + X
  }
  __syncthreads();
  float fac = 1.0f;
#pragma unroll 1
  for (int i = 2; i <= 5; ++i) {        // order-5 Taylor (matches reference)
    mm32(P, XD, T, lane, 1.0f);
    copy_m(P, T, lane); __syncthreads();
    fac *= (float)i;
    add_scaled(OUT, P, 1.0f / fac, lane);
  }
  mm32(S, OUT, T, lane, 1.0f);          // Gs * expm
  mm32(T, S, OUT, lane, 1.0f);          // (Gs*expm) * Gs
  for (int t = lane; t < DD; t += 32) {
    const int r = t >> 5, c = t & 31;
    float v = 0.5f * (OUT[t] + OUT[c * DIM + r]);
    if (r == c) v += JITTER;
    G[(size_t)b * DD + t] = v;
  }
}

__global__ void check_kernel(const float* nrm, int* done) {
  if (*nrm < TOL) *done = 1;
}

__global__ void copy_kernel(const float* __restrict__ G, float* __restrict__ out) {
  const int t = blockIdx.x * blockDim.x + threadIdx.x;
  if (t < BATCH * DD) out[t] = G[t];
}

// ---------------------------------------------------------------------------
extern "C" void kernel_launch(void* const* d_in, const int* in_sizes, int n_in,
                              void* d_out, int out_size, void* d_ws, size_t ws_size,
                              hipStream_t stream) {
  const float* X = (const float*)d_in[0];
  float* out = (float*)d_out;

  float* G     = (float*)d_ws;          // 64*1024
  float* Gs    = G     + BATCH * DD;
  float* Gis   = Gs    + BATCH * DD;
  float* Delta = Gis   + BATCH * DD;
  float* nrm   = Delta + BATCH * DD;
  int*   done  = (int*)(nrm + 1);

  init_kernel<<<1, 1, 0, stream>>>(nrm, done);
  mean_kernel<<<BATCH, 256, 0, stream>>>(X, G);

  for (int it = 0; it < 10; ++it) {
    sqrt_invsqrt_kernel<<<BATCH, 32, 0, stream>>>(G, Gs, Gis);
    zero_kernel<<<(BATCH * DD + 255) / 256, 256, 0, stream>>>(Delta, nrm);
    log_accum_kernel<<<BATCH * NMAT, 32, 0, stream>>>(X, Gis, Delta);
    norm_kernel<<<BATCH, 32, 0, stream>>>(Delta, nrm);
    update_kernel<<<BATCH, 32, 0, stream>>>(G, Gs, Delta, done);   // uses done from prev iters
    check_kernel<<<1, 1, 0, stream>>>(nrm, done);                  // done |= (norm < TOL)
  }

  copy_kernel<<<(BATCH * DD + 255) / 256, 256, 0, stream>>>(G, out);
}